// Spmm_GAT_29850022707670
// MI455X (gfx1250) — compile-verified
//
#include <hip/hip_runtime.h>
#include <math.h>

typedef __attribute__((ext_vector_type(2))) float v2f;
typedef __attribute__((ext_vector_type(8))) float v8f;

#define NEG_SLOPE 0.2f

// ---------------------------------------------------------------- utilities

__device__ __forceinline__ float waveReduceSum(float v) {
#pragma unroll
  for (int m = 16; m >= 1; m >>= 1) v += __shfl_xor(v, m, 32);
  return v;
}

__device__ __forceinline__ float waveReduceMax(float v) {
#pragma unroll
  for (int m = 16; m >= 1; m >>= 1) v = fmaxf(v, __shfl_xor(v, m, 32));
  return v;
}

__device__ __forceinline__ float leaky(float l) {
  return (l > 0.f) ? l : NEG_SLOPE * l;
}

__global__ void fill_i32(int* __restrict__ p, long n, int v) {
  long i = (long)blockIdx.x * blockDim.x + threadIdx.x;
  if (i < n) p[i] = v;
}

// ---------------------------------------------------------------- CSR build
// Sort edges by destination so edge-softmax + aggregation need no fp atomics.

__global__ void csr_degree(const int* __restrict__ dst, int* __restrict__ deg, long E) {
  long e = (long)blockIdx.x * blockDim.x + threadIdx.x;
  if (e < E) atomicAdd(&deg[dst[e]], 1);
}

// Single-block chunked exclusive scan: rowstart[0..N] and cursor copy.
__global__ void csr_scan(const int* __restrict__ deg, int* __restrict__ rowstart,
                         int* __restrict__ cursor, int N) {
  __shared__ int sh[1024];
  __shared__ int carry;
  const int tid = threadIdx.x;
  if (tid == 0) carry = 0;
  __syncthreads();
  for (int base = 0; base < N; base += 1024) {
    int v = (base + tid < N) ? deg[base + tid] : 0;
    sh[tid] = v;
    __syncthreads();
    for (int off = 1; off < 1024; off <<= 1) {
      int t = (tid >= off) ? sh[tid - off] : 0;
      __syncthreads();
      sh[tid] += t;
      __syncthreads();
    }
    int excl = carry + sh[tid] - v;   // exclusive prefix
    if (base + tid < N) { rowstart[base + tid] = excl; cursor[base + tid] = excl; }
    __syncthreads();
    if (tid == 0) carry += sh[1023];
    __syncthreads();
  }
  if (tid == 0) rowstart[N] = carry;
}

// Scatter the source index of each edge into its dst-sorted slot.
__global__ void csr_scatter(const int* __restrict__ src, const int* __restrict__ dst,
                            int* __restrict__ cursor, int* __restrict__ srcsorted, long E) {
  long e = (long)blockIdx.x * blockDim.x + threadIdx.x;
  if (e >= E) return;
  int pos = atomicAdd(&cursor[dst[e]], 1);
  srcsorted[pos] = src[e];
}

// ---------------------------------------------------------------- dense GEMM
// C[M,Nout] = A[M,K] x B[K,Nout], fp32 row-major, V_WMMA_F32_16X16X4_F32.
// One wave computes a 16x64 tile: 1 A-fragment feeds 4 WMMAs per K-step.
__global__ void gemm_wmma_f32(const float* __restrict__ A,
                              const float* __restrict__ B,
                              float* __restrict__ C,
                              int M, int K, int Nout) {
  const int lane    = threadIdx.x & 31;
  const int wave    = threadIdx.x >> 5;
  const int rowTile = blockIdx.x * (blockDim.x >> 5) + wave;  // 16-row tile id
  if (rowTile * 16 >= M) return;                              // wave-uniform
  const int row0    = rowTile * 16;
  const int colBase = blockIdx.y * 64;

  // A 16x4 f32 fragment: lanes 0-15 hold K=k+0,k+1 for row=lane;
  // lanes 16-31 hold K=k+2,k+3 for row=lane-16.
  const int aRow  = row0 + (lane & 15);
  const int kHalf = (lane >> 4) << 1;  // 0 or 2
  const int bCol  = colBase + (lane & 15);

  v8f acc0 = {}, acc1 = {}, acc2 = {}, acc3 = {};

  for (int k = 0; k < K; k += 4) {
    v2f a;
    const float* Ap = A + (size_t)aRow * K + (k + kHalf);
    a.x = Ap[0];
    a.y = Ap[1];

    const float* Bk0 = B + (size_t)(k + kHalf) * Nout + bCol;
    const float* Bk1 = Bk0 + Nout;
    v2f b0; b0.x = Bk0[0];  b0.y = Bk1[0];
    v2f b1; b1.x = Bk0[16]; b1.y = Bk1[16];
    v2f b2; b2.x = Bk0[32]; b2.y = Bk1[32];
    v2f b3; b3.x = Bk0[48]; b3.y = Bk1[48];

    acc0 = __builtin_amdgcn_wmma_f32_16x16x4_f32(false, a, false, b0, (short)0, acc0, false, false);
    acc1 = __builtin_amdgcn_wmma_f32_16x16x4_f32(false, a, false, b1, (short)0, acc1, false, false);
    acc2 = __builtin_amdgcn_wmma_f32_16x16x4_f32(false, a, false, b2, (short)0, acc2, false, false);
    acc3 = __builtin_amdgcn_wmma_f32_16x16x4_f32(false, a, false, b3, (short)0, acc3, false, false);
  }

  // C/D layout: VGPR r -> row r (lanes 0-15) / row r+8 (lanes 16-31), col = lane&15
  const int cRowBase = row0 + ((lane >> 4) << 3);
  const int cCol     = colBase + (lane & 15);
#pragma unroll
  for (int r = 0; r < 8; ++r) {
    float* Cp = C + (size_t)(cRowBase + r) * Nout + cCol;
    Cp[0]  = acc0[r];
    Cp[16] = acc1[r];
    Cp[32] = acc2[r];
    Cp[48] = acc3[r];
  }
}

// ------------------------------------------------- per-node attention logits
// e_src[n,h] = sum_c h[n,h,c]*a_src[h,c] ; e_dst likewise. One wave per (n,h).
__global__ void node_logits(const float* __restrict__ h,
                            const float* __restrict__ a_src,
                            const float* __restrict__ a_dst,
                            float* __restrict__ e_src,
                            float* __restrict__ e_dst,
                            int N, int H, int C) {
  const int lane = threadIdx.x & 31;
  const int idx  = blockIdx.x * (blockDim.x >> 5) + (threadIdx.x >> 5);
  if (idx >= N * H) return;
  const int hh = idx % H;
  const float* hp  = h + (size_t)idx * C;   // [N*H, C] contiguous
  const float* asp = a_src + hh * C;
  const float* adp = a_dst + hh * C;
  float ss = 0.f, sd = 0.f;
  for (int c = lane; c < C; c += 32) {
    float v = hp[c];
    ss += v * asp[c];
    sd += v * adp[c];
  }
  ss = waveReduceSum(ss);
  sd = waveReduceSum(sd);
  if (lane == 0) { e_src[idx] = ss; e_dst[idx] = sd; }
}

// ---------------------------------------- fused edge softmax + aggregation
// One wave per destination node, walking its dst-sorted edge segment.
// Pass 1: segment max (lanes stride edges).  Pass 2: exp-sum.
// Pass 3: alpha-weighted gather-accumulate; each lane owns a CONTIGUOUS block
//         of J = H*C/32 channels so gathers/stores are 128-bit, and the block
//         lies inside one head (C % J == 0) so one alpha per lane suffices.
// No atomics, no init fills; empty nodes write zeros (matches reference).
template <int H, int C>
__global__ void gat_softmax_aggregate(const int* __restrict__ rowstart,
                                      const int* __restrict__ srcsorted,
                                      const float* __restrict__ e_src,
                                      const float* __restrict__ e_dst,
                                      const float* __restrict__ hfeat,
                                      float* __restrict__ agg, int N) {
  constexpr int HC = H * C;
  constexpr int J  = HC / 32;          // channels per lane (8 or 2)
  const int lane = threadIdx.x & 31;
  int n = blockIdx.x * (blockDim.x >> 5) + (threadIdx.x >> 5);
  if (n >= N) return;
  const int beg = rowstart[n], end = rowstart[n + 1];

  float edv[H];
#pragma unroll
  for (int h = 0; h < H; ++h) edv[h] = e_dst[(size_t)n * H + h];

  // ---- pass 1: segment max per head (lanes stride edges)
  float mx[H];
#pragma unroll
  for (int h = 0; h < H; ++h) mx[h] = -__builtin_inff();
  for (int j = beg + lane; j < end; j += 32) {
    int s = srcsorted[j];
    if constexpr (H == 4) {
      float4 t = reinterpret_cast<const float4*>(e_src)[s];
      mx[0] = fmaxf(mx[0], leaky(t.x + edv[0]));
      mx[1] = fmaxf(mx[1], leaky(t.y + edv[1]));
      mx[2] = fmaxf(mx[2], leaky(t.z + edv[2]));
      mx[3] = fmaxf(mx[3], leaky(t.w + edv[3]));
    } else {
#pragma unroll
      for (int h = 0; h < H; ++h)
        mx[h] = fmaxf(mx[h], leaky(e_src[(size_t)s * H + h] + edv[h]));
    }
  }
#pragma unroll
  for (int h = 0; h < H; ++h) mx[h] = waveReduceMax(mx[h]);

  // ---- pass 2: exp-sum per head
  float rdn[H];
  {
    float dn[H];
#pragma unroll
    for (int h = 0; h < H; ++h) dn[h] = 0.f;
    for (int j = beg + lane; j < end; j += 32) {
      int s = srcsorted[j];
      if constexpr (H == 4) {
        float4 t = reinterpret_cast<const float4*>(e_src)[s];
        dn[0] += expf(leaky(t.x + edv[0]) - mx[0]);
        dn[1] += expf(leaky(t.y + edv[1]) - mx[1]);
        dn[2] += expf(leaky(t.z + edv[2]) - mx[2]);
        dn[3] += expf(leaky(t.w + edv[3]) - mx[3]);
      } else {
#pragma unroll
        for (int h = 0; h < H; ++h)
          dn[h] += expf(leaky(e_src[(size_t)s * H + h] + edv[h]) - mx[h]);
      }
    }
#pragma unroll
    for (int h = 0; h < H; ++h) rdn[h] = 1.f / (waveReduceSum(dn[h]) + 1e-16f);
  }

  // ---- pass 3: alpha-weighted gather-accumulate, 128-bit vector traffic
  const int   myh  = (lane * J) / C;      // single head covers this lane's block
  const float myed = edv[myh];
  const float mymx = mx[myh];
  const float myrd = rdn[myh];

  if constexpr (J % 4 == 0) {
    constexpr int V = J / 4;              // float4s per lane
    float4 acc[V];
#pragma unroll
    for (int v = 0; v < V; ++v) acc[v] = make_float4(0.f, 0.f, 0.f, 0.f);
    for (int j = beg; j < end; ++j) {
      int s = srcsorted[j];               // wave-uniform
      float alpha = expf(leaky(e_src[(size_t)s * H + myh] + myed) - mymx) * myrd;
      const float4* hp = reinterpret_cast<const float4*>(hfeat + (size_t)s * HC) + lane * V;
#pragma unroll
      for (int v = 0; v < V; ++v) {
        float4 t = hp[v];
        acc[v].x += alpha * t.x;
        acc[v].y += alpha * t.y;
        acc[v].z += alpha * t.z;
        acc[v].w += alpha * t.w;
      }
    }
    float4* op = reinterpret_cast<float4*>(agg + (size_t)n * HC) + lane * V;
#pragma unroll
    for (int v = 0; v < V; ++v) op[v] = acc[v];
  } else {                                // J == 2
    float2 acc = make_float2(0.f, 0.f);
    for (int j = beg; j < end; ++j) {
      int s = srcsorted[j];
      float alpha = expf(leaky(e_src[(size_t)s * H + myh] + myed) - mymx) * myrd;
      float2 t = reinterpret_cast<const float2*>(hfeat + (size_t)s * HC)[lane];
      acc.x += alpha * t.x;
      acc.y += alpha * t.y;
    }
    reinterpret_cast<float2*>(agg + (size_t)n * HC)[lane] = acc;
  }
}

// ------------------------------------------------- bias + ELU (in place OK)
__global__ void bias_elu(const float* __restrict__ agg,
                         const float* __restrict__ b,
                         float* __restrict__ out, long total, int F) {
  long i = (long)blockIdx.x * blockDim.x + threadIdx.x;
  if (i >= total) return;
  float v = agg[i] + b[(int)(i % F)];
  out[i] = (v > 0.f) ? v : (expf(v) - 1.f);
}

// -------------------------------------- final: bias + log_softmax over C=64
__global__ void bias_logsoftmax64(const float* __restrict__ agg,
                                  const float* __restrict__ b,
                                  float* __restrict__ out, int N) {
  const int lane = threadIdx.x & 31;
  int n = blockIdx.x * (blockDim.x >> 5) + (threadIdx.x >> 5);
  if (n >= N) return;
  float v0 = agg[(size_t)n * 64 + lane]      + b[lane];
  float v1 = agg[(size_t)n * 64 + lane + 32] + b[lane + 32];
  float mx = waveReduceMax(fmaxf(v0, v1));
  float se = waveReduceSum(expf(v0 - mx) + expf(v1 - mx));
  float lse = mx + logf(se);
  out[(size_t)n * 64 + lane]      = v0 - lse;
  out[(size_t)n * 64 + lane + 32] = v1 - lse;
}

// ---------------------------------------------------------------- launcher

static inline int ceil_div_l(long a, int b) { return (int)((a + b - 1) / b); }

extern "C" void kernel_launch(void* const* d_in, const int* in_sizes, int n_in,
                              void* d_out, int out_size, void* d_ws, size_t ws_size,
                              hipStream_t stream) {
  (void)n_in; (void)out_size; (void)ws_size;

  const float* x    = (const float*)d_in[0];
  const int*   edge = (const int*)  d_in[1];
  const float* W1   = (const float*)d_in[2];
  const float* a1s  = (const float*)d_in[3];
  const float* a1d  = (const float*)d_in[4];
  const float* b1   = (const float*)d_in[5];
  const float* W2   = (const float*)d_in[6];
  const float* a2s  = (const float*)d_in[7];
  const float* a2d  = (const float*)d_in[8];
  const float* b2   = (const float*)d_in[9];

  const int IN_DIM = 256;
  const int H1 = 4, C1 = 64, HC1 = H1 * C1;
  const int H2 = 1, C2 = 64, HC2 = H2 * C2;
  const int  N = in_sizes[0] / IN_DIM;
  const long E = (long)in_sizes[1] / 2;
  const int* srcIdx = edge;
  const int* dstIdx = edge + E;

  // ----- carve workspace (all 4-byte elements; big arrays first keep
  //       16-byte alignment for the float4 paths)
  float* ws = (float*)d_ws;
  size_t o = 0;
  float* h1   = ws + o; o += (size_t)N * HC1;   // layer-1 features (x@W1)
  float* agg1 = ws + o; o += (size_t)N * HC1;   // aggregate, then ELU in place
  float* h2   = ws + o; o += (size_t)N * HC2;   // layer-2 features
  float* agg2 = ws + o; o += (size_t)N * HC2;
  float* es1  = ws + o; o += (size_t)N * H1;
  float* ed1  = ws + o; o += (size_t)N * H1;
  float* es2  = ws + o; o += (size_t)N * H2;
  float* ed2  = ws + o; o += (size_t)N * H2;
  int* deg       = (int*)(ws + o); o += (size_t)N;
  int* cursor    = (int*)(ws + o); o += (size_t)N;
  int* rowstart  = (int*)(ws + o); o += (size_t)N + 1;
  int* srcsorted = (int*)(ws + o); o += (size_t)E;

  const int TPB = 256;
  const int WPB = TPB >> 5;  // 8 waves per block

  // ================= CSR build (shared by both layers) =================
  fill_i32<<<ceil_div_l((long)N, TPB), TPB, 0, stream>>>(deg, (long)N, 0);
  csr_degree<<<ceil_div_l(E, TPB), TPB, 0, stream>>>(dstIdx, deg, E);
  csr_scan<<<1, 1024, 0, stream>>>(deg, rowstart, cursor, N);
  csr_scatter<<<ceil_div_l(E, TPB), TPB, 0, stream>>>(srcIdx, dstIdx, cursor, srcsorted, E);

  // ================= layer 1 =================
  {  // h1 = x @ W1   (M=N, K=256, Nout=256)
    dim3 grid(ceil_div_l((N + 15) / 16, WPB), HC1 / 64);
    gemm_wmma_f32<<<grid, TPB, 0, stream>>>(x, W1, h1, N, IN_DIM, HC1);
  }
  node_logits<<<ceil_div_l((long)N * H1, WPB), TPB, 0, stream>>>(
      h1, a1s, a1d, es1, ed1, N, H1, C1);
  gat_softmax_aggregate<4, 64><<<ceil_div_l((long)N, WPB), TPB, 0, stream>>>(
      rowstart, srcsorted, es1, ed1, h1, agg1, N);
  bias_elu<<<ceil_div_l((long)N * HC1, TPB), TPB, 0, stream>>>(
      agg1, b1, agg1, (long)N * HC1, HC1);   // ELU in place -> layer-2 input

  // ================= layer 2 =================
  {  // h2 = elu(agg1+b1) @ W2  (M=N, K=256, Nout=64)
    dim3 grid(ceil_div_l((N + 15) / 16, WPB), HC2 / 64);
    gemm_wmma_f32<<<grid, TPB, 0, stream>>>(agg1, W2, h2, N, HC1, HC2);
  }
  node_logits<<<ceil_div_l((long)N * H2, WPB), TPB, 0, stream>>>(
      h2, a2s, a2d, es2, ed2, N, H2, C2);
  gat_softmax_aggregate<1, 64><<<ceil_div_l((long)N, WPB), TPB, 0, stream>>>(
      rowstart, srcsorted, es2, ed2, h2, agg2, N);

  // mean over H2=1 heads is identity; final bias + log_softmax -> d_out
  bias_logsoftmax64<<<ceil_div_l((long)N, WPB), TPB, 0, stream>>>(
      agg2, b2, (float*)d_out, N);
}